// MoAT_69097433858279
// MI455X (gfx1250) — compile-verified
//
#include <hip/hip_runtime.h>
#include <math.h>

#define NDIM 512
#define BDIM 256
#define EPSF 1e-7f

#define NB   32          // LU panel width
#define LPST 33          // LDS stride for panel (bank-conflict padding)
#define UPST 480         // LDS stride for U12 row-panel (max trailing cols)

typedef __attribute__((ext_vector_type(2))) float v2f;
typedef __attribute__((ext_vector_type(8))) float v8f;

// ---------------- V = softmax(V_compress, axis=1)  (512 rows x 4) ----------------
__global__ void k_softmaxV(const float* __restrict__ Vc, float* __restrict__ Vv) {
    int i = blockIdx.x * blockDim.x + threadIdx.x;
    if (i >= NDIM) return;
    float v0 = Vc[i*4+0], v1 = Vc[i*4+1], v2 = Vc[i*4+2], v3 = Vc[i*4+3];
    float m = fmaxf(fmaxf(v0, v1), fmaxf(v2, v3));
    float e0 = expf(v0-m), e1 = expf(v1-m), e2 = expf(v2-m), e3 = expf(v3-m);
    float inv = 1.0f / (e0+e1+e2+e3);
    Vv[i*4+0] = e0*inv; Vv[i*4+1] = e1*inv; Vv[i*4+2] = e2*inv; Vv[i*4+3] = e3*inv;
}

// ---------------- Ws = sym(tril(sigmoid(W),-1)) ----------------
__global__ void k_Ws(const float* __restrict__ W, float* __restrict__ Ws) {
    int idx = blockIdx.x * 256 + threadIdx.x;      // over N*N
    int i = idx >> 9, j = idx & (NDIM-1);
    float v = 0.0f;
    if (i > j)      v = 1.0f / (1.0f + expf(-W[(size_t)i*NDIM + j]));
    else if (i < j) v = 1.0f / (1.0f + expf(-W[(size_t)j*NDIM + i]));
    Ws[idx] = v;
}

__global__ void k_rowsumWs(const float* __restrict__ Ws, float* __restrict__ rs) {
    int i = blockIdx.x * blockDim.x + threadIdx.x;
    if (i >= NDIM) return;
    float s = 0.0f;
    for (int j = 0; j < NDIM; ++j) s += Ws[(size_t)i*NDIM + j];
    rs[i] = s;
}

// M0 = L_0[1:,1:] + EPS, padded to 512x512 with unit last row/col
__global__ void k_M0(const float* __restrict__ Ws, const float* __restrict__ rs,
                     float* __restrict__ M0) {
    int idx = blockIdx.x * 256 + threadIdx.x;      // over N*N
    int r = idx >> 9, c = idx & (NDIM-1);
    float v;
    if (r == NDIM-1 || c == NDIM-1) {
        v = (r == c) ? 1.0f : 0.0f;
    } else {
        int i = r + 1, j = c + 1;
        v = -Ws[(size_t)i*NDIM + j] + ((i == j) ? rs[i] : 0.0f) + EPSF;
    }
    M0[idx] = v;
}

// ---------------- Sinkhorn: per (i,j) independent 4x4 problem ----------------
__global__ void k_sinkhorn(const float* __restrict__ Ec, const float* __restrict__ Vv,
                           float* __restrict__ E) {
    int idx = blockIdx.x * 256 + threadIdx.x;      // over N*N
    int i = idx >> 9, j = idx & (NDIM-1);
    float e[16];
    float s = 0.0f;
    #pragma unroll
    for (int k = 0; k < 16; ++k) { e[k] = expf(Ec[(size_t)idx*16 + k]); s += e[k]; }
    float inv = 1.0f / s;
    #pragma unroll
    for (int k = 0; k < 16; ++k) e[k] *= inv;
    float Vi[4], Vj[4];
    #pragma unroll
    for (int a = 0; a < 4; ++a) { Vi[a] = Vv[i*4+a]; Vj[a] = Vv[j*4+a]; }
    #pragma unroll
    for (int it = 0; it < 2; ++it) {
        #pragma unroll
        for (int a = 0; a < 4; ++a) {
            float rsum = e[4*a] + e[4*a+1] + e[4*a+2] + e[4*a+3];
            float sc = Vi[a] / rsum;
            e[4*a] *= sc; e[4*a+1] *= sc; e[4*a+2] *= sc; e[4*a+3] *= sc;
        }
        #pragma unroll
        for (int b = 0; b < 4; ++b) {
            float csum = e[b] + e[4+b] + e[8+b] + e[12+b];
            float sc = Vj[b] / csum;
            e[b] *= sc; e[4+b] *= sc; e[8+b] *= sc; e[12+b] *= sc;
        }
        float tot = 0.0f;
        #pragma unroll
        for (int k = 0; k < 16; ++k) tot += e[k];
        float it_inv = 1.0f / tot;
        #pragma unroll
        for (int k = 0; k < 16; ++k) e[k] *= it_inv;
    }
    float mask = (i == j) ? 0.0f : 1.0f;
    #pragma unroll
    for (int k = 0; k < 16; ++k)
        E[(size_t)idx*16 + k] = fminf(fmaxf(e[k]*mask, 0.0f), 1.0f);
}

// ---------------- Pr[b,i] = V[i, x[b,i]]  and  logPr sums ----------------
__global__ void k_Pr(const int* __restrict__ x, const float* __restrict__ Vv,
                     float* __restrict__ Pr) {
    int idx = blockIdx.x * 256 + threadIdx.x;      // over B*N
    Pr[idx] = Vv[(idx & (NDIM-1))*4 + x[idx]];
}

__global__ void k_lps(const float* __restrict__ Pr, float* __restrict__ lps) {
    int b = blockIdx.x, tid = threadIdx.x;
    __shared__ float red[256];
    float s = logf(Pr[(size_t)b*NDIM + tid] + EPSF)
            + logf(Pr[(size_t)b*NDIM + tid + 256] + EPSF);
    red[tid] = s; __syncthreads();
    for (int off = 128; off > 0; off >>= 1) {
        if (tid < off) red[tid] += red[tid+off];
        __syncthreads();
    }
    if (tid == 0) lps[b] = red[0];
}

// ---------------- Build Lap[b][1:,1:]+EPS (padded 512x512), one block per row --------
__global__ void __launch_bounds__(256)
k_buildM(const int* __restrict__ x, const float* __restrict__ Ws,
         const float* __restrict__ E, const float* __restrict__ Pr,
         float* __restrict__ M, int b0) {
    int r  = blockIdx.x;
    int lb = blockIdx.y;
    int b  = b0 + lb;
    float* Mb = M + (size_t)lb * NDIM * NDIM;
    int tid = threadIdx.x;
    if (r == NDIM-1) {                      // pad row: e_511
        for (int c = tid; c < NDIM; c += 256)
            Mb[(size_t)r*NDIM + c] = (c == NDIM-1) ? 1.0f : 0.0f;
        return;
    }
    __shared__ float red[256];
    int   i = r + 1;
    int   xi = x[(size_t)b*NDIM + i];
    float invpri = 1.0f / Pr[(size_t)b*NDIM + i];
    float s = 0.0f;
    for (int j = tid; j < NDIM; j += 256) {
        int   xj  = x[(size_t)b*NDIM + j];
        float prj = Pr[(size_t)b*NDIM + j];
        float ws  = Ws[(size_t)i*NDIM + j];
        float e   = E[((size_t)i*NDIM + j)*16 + xi*4 + xj]; // zero at i==j (masked)
        float wp  = ws * e * invpri / prj;
        s += wp;
        if (j >= 1) Mb[(size_t)r*NDIM + (j-1)] = EPSF - wp;
    }
    if (tid == 0) Mb[(size_t)r*NDIM + (NDIM-1)] = 0.0f;    // pad col
    red[tid] = s; __syncthreads();
    for (int off = 128; off > 0; off >>= 1) {
        if (tid < off) red[tid] += red[tid+off];
        __syncthreads();
    }
    if (tid == 0) Mb[(size_t)r*NDIM + r] = red[0] + EPSF;  // diag = rowsum(WP)+EPS
}

// ---------------- Batched LU (no pivoting) + log|det|, WMMA trailing update ---------
// One 512-thread (16-wave) workgroup per 512x512 matrix; NB=32 panels.
// Trailing GEMM: exact 2x2 register-blocked 16x16 tiles per wave (trailing width is
// always a multiple of 32 => mt even => NO remainder guards, WMMA unconditional,
// EXEC all-ones). Wave id is readfirstlane'd so all tile-loop control is scalar.
// U12 stored NEGATED in LDS => hot loop is pure ds_load + v_wmma.
__global__ void __launch_bounds__(512)
k_lu(float* __restrict__ Mbase, float* __restrict__ ldOut) {
    float* A = Mbase + (size_t)blockIdx.x * NDIM * NDIM;
    __shared__ float Lp[NDIM * LPST];   // current panel (rows k0..511, NB cols)
    __shared__ float Up[NB * UPST];     // -U12 row-panel (NB x up-to-480)
    __shared__ float sld;
    const int tid = threadIdx.x;
    const int nthr = 512;
    if (tid == 0) sld = 0.0f;

    for (int k0 = 0; k0 < NDIM; k0 += NB) {
        const int m = NDIM - k0;
        // ---- stage panel into LDS
        for (int idx = tid; idx < m*NB; idx += nthr) {
            int rel = idx >> 5, jj = idx & (NB-1);
            Lp[rel*LPST + jj] = A[(size_t)(k0+rel)*NDIM + k0 + jj];
        }
        __syncthreads();
        // ---- unblocked panel factorization (no pivoting; diag-dominant input)
        for (int j = 0; j < NB; ++j) {
            float inv = 1.0f / Lp[j*LPST + j];
            for (int rel = j + 1 + tid; rel < m; rel += nthr) {
                float lij = Lp[rel*LPST + j] * inv;
                Lp[rel*LPST + j] = lij;
                for (int jj = j + 1; jj < NB; ++jj)
                    Lp[rel*LPST + jj] -= lij * Lp[j*LPST + jj];
            }
            __syncthreads();
        }
        if (tid == 0) {
            float s = 0.0f;
            for (int j = 0; j < NB; ++j) s += logf(fabsf(Lp[j*LPST + j]));
            sld += s;
        }
        // ---- U12 = L11^{-1} * A12 (NB-deep triangular solve per column) -> -U12 in LDS
        for (int cc = tid; cc < m - NB; cc += nthr) {
            int c = k0 + NB + cc;
            float u[NB];
            #pragma unroll
            for (int rr = 0; rr < NB; ++rr) u[rr] = A[(size_t)(k0+rr)*NDIM + c];
            #pragma unroll
            for (int rr = 1; rr < NB; ++rr) {
                float ur = u[rr];
                for (int t = 0; t < rr; ++t) ur -= Lp[rr*LPST + t] * u[t];
                u[rr] = ur;
            }
            #pragma unroll
            for (int rr = 0; rr < NB; ++rr) Up[rr*UPST + cc] = -u[rr];  // negated!
        }
        __syncthreads();
        // ---- trailing update A22 += L21 * (-U12) via f32 WMMA, 2x2 tile blocks
        if (m > NB) {
            const int mt   = (m - NB) >> 4;   // 16x16 tiles per dim (always EVEN)
            const int nbt  = mt >> 1;         // 2x2 tile blocks per dim (exact)
            const int nblk = nbt * nbt;
            const int w    = __builtin_amdgcn_readfirstlane(tid >> 5); // scalar wave id
            const int lane = tid & 31;
            const int lm   = lane & 15;
            const int hi   = lane >> 4;
            for (int t = w; t < nblk; t += 16) {      // scalar loop control
                const int bi = t / nbt, bj = t - bi*nbt;
                const int i0 = k0 + NB + (bi << 5);   // top row of 2x2 block
                const int c0 = k0 + NB + (bj << 5);   // left col of 2x2 block
                v8f c[2][2];
                #pragma unroll
                for (int di = 0; di < 2; ++di)
                  #pragma unroll
                  for (int dj = 0; dj < 2; ++dj)
                    #pragma unroll
                    for (int v = 0; v < 8; ++v)
                        c[di][dj][v] = A[(size_t)(i0 + di*16 + v + hi*8)*NDIM
                                         + c0 + dj*16 + lm];
                #pragma unroll
                for (int kk = 0; kk < NB/4; ++kk) {
                    const int kb = (kk << 2) + (hi << 1);
                    v2f a[2], bf[2];
                    #pragma unroll
                    for (int di = 0; di < 2; ++di) {
                        const int arow = (i0 - k0) + di*16 + lm;   // row rel to k0
                        a[di].x = Lp[arow*LPST + kb];
                        a[di].y = Lp[arow*LPST + kb + 1];
                    }
                    #pragma unroll
                    for (int dj = 0; dj < 2; ++dj) {
                        const int cc0 = (c0 - k0 - NB) + dj*16 + lm;
                        bf[dj].x = Up[kb      *UPST + cc0];
                        bf[dj].y = Up[(kb + 1)*UPST + cc0];
                    }
                    #pragma unroll
                    for (int di = 0; di < 2; ++di)
                      #pragma unroll
                      for (int dj = 0; dj < 2; ++dj)
                        c[di][dj] = __builtin_amdgcn_wmma_f32_16x16x4_f32(
                            false, a[di], false, bf[dj], (short)0, c[di][dj],
                            false, false);
                }
                #pragma unroll
                for (int di = 0; di < 2; ++di)
                  #pragma unroll
                  for (int dj = 0; dj < 2; ++dj)
                    #pragma unroll
                    for (int v = 0; v < 8; ++v)
                        A[(size_t)(i0 + di*16 + v + hi*8)*NDIM
                          + c0 + dj*16 + lm] = c[di][dj][v];
            }
        }
        __syncthreads();
    }
    if (tid == 0) ldOut[blockIdx.x] = sld;
}

// ---------------- y[b] = lps[b] + ld[b] - ld0 ----------------
__global__ void k_y(const float* __restrict__ lps, const float* __restrict__ ld,
                    float* __restrict__ y) {
    int b = threadIdx.x;
    y[b] = lps[b] + ld[b] - ld[256];
}

extern "C" void kernel_launch(void* const* d_in, const int* in_sizes, int n_in,
                              void* d_out, int out_size, void* d_ws, size_t ws_size,
                              hipStream_t stream) {
    (void)in_sizes; (void)n_in; (void)out_size;
    const int*   x  = (const int*)  d_in[0];
    const float* W  = (const float*)d_in[1];
    const float* Vc = (const float*)d_in[2];
    const float* Ec = (const float*)d_in[3];
    float* y = (float*)d_out;

    float* p = (float*)d_ws;
    auto alloc = [&](size_t nfloats) {
        float* r = p;
        p += (nfloats + 63) & ~(size_t)63;   // 256B-align each region
        return r;
    };
    float* Vv   = alloc((size_t)NDIM*4);
    float* rsWs = alloc(NDIM);
    float* Ws   = alloc((size_t)NDIM*NDIM);
    float* Pr   = alloc((size_t)BDIM*NDIM);
    float* lps  = alloc(BDIM);
    float* ld   = alloc(BDIM + 1);
    float* M0   = alloc((size_t)NDIM*NDIM);
    float* E    = alloc((size_t)NDIM*NDIM*16);
    size_t used  = (size_t)((char*)p - (char*)d_ws);
    size_t avail = ws_size > used ? ws_size - used : 0;
    int nbMax = (int)(avail / ((size_t)NDIM*NDIM*sizeof(float)));
    int CH = nbMax < 1 ? 1 : (nbMax > 128 ? 128 : nbMax); // 128MB chunk: L2-resident
    float* M = p;

    k_softmaxV<<<2, 256, 0, stream>>>(Vc, Vv);
    k_Ws      <<<(NDIM*NDIM)/256, 256, 0, stream>>>(W, Ws);
    k_rowsumWs<<<2, 256, 0, stream>>>(Ws, rsWs);
    k_M0      <<<(NDIM*NDIM)/256, 256, 0, stream>>>(Ws, rsWs, M0);
    k_sinkhorn<<<(NDIM*NDIM)/256, 256, 0, stream>>>(Ec, Vv, E);
    k_Pr      <<<(BDIM*NDIM)/256, 256, 0, stream>>>(x, Vv, Pr);
    k_lps     <<<BDIM, 256, 0, stream>>>(Pr, lps);
    k_lu      <<<1, 512, 0, stream>>>(M0, ld + 256);
    for (int b0 = 0; b0 < BDIM; b0 += CH) {
        int nb = (BDIM - b0) < CH ? (BDIM - b0) : CH;
        k_buildM<<<dim3(NDIM, nb), 256, 0, stream>>>(x, Ws, E, Pr, M, b0);
        k_lu    <<<nb, 512, 0, stream>>>(M, ld + b0);
    }
    k_y<<<1, 256, 0, stream>>>(lps, ld, y);
}